// GMMLoss_3779571220770
// MI455X (gfx1250) — compile-verified
//
#include <hip/hip_runtime.h>
#include <math.h>

// GMM NLL loss: B=4, N=4096 preds, M=4096 targets.
// Rank-6 bilinear score matrix -> padded K=8 FP32 WMMA (2x 16x16x4 chained),
// fused with online logsumexp over N (flash-attn style).
//
// v3:
//  - raw v_exp_f32 / v_log_f32 via __builtin_amdgcn_exp2f / _logf (no OCML
//    subnormal-guard code: args are <=0 deltas where flush-to-zero is the
//    mathematically desired LSE behavior).
//  - F log2(e)-prescaled (scores natively log2-domain, no per-element mul).
//  - F/G in WMMA-fragment-permuted order: one global_load_b128 per A tile.
//  - #pragma unroll 2 to restore dual-tile WMMA/load pipelining.

#define BATCH 4
#define NPRED 4096
#define MTGT  4096
#define EPS_F 1e-8f
#define LOG_2PI_F 1.8378762f        /* ln(2*3.14159) */
#define LOG2E_F   1.4426950408889634f
#define LN2_F     0.6931471805599453f

typedef __attribute__((ext_vector_type(2))) float v2f;
typedef __attribute__((ext_vector_type(4))) float v4f;
typedef __attribute__((ext_vector_type(8))) float v8f;

__device__ __forceinline__ float fast_exp2(float x) {
    return __builtin_amdgcn_exp2f(x);     // bare v_exp_f32
}
__device__ __forceinline__ float fast_log2(float x) {
    return __builtin_amdgcn_logf(x);      // bare v_log_f32
}

// ---------------------------------------------------------------------------
// Kernel 1: build features, permuted order [k0,k1,k4,k5, k2,k3,k6,k7].
//   log2_prob[n,m] = dot(F[n], G[m])   (F carries the log2e scaling)
//   k0=c0, k1=-a, k2=2a*px, k3=2a*py, k4=-bz, k5=2bz*pz
//   g0=1,  g1=|t_xy|^2, g2=tx, g3=ty, g4=tz^2, g5=tz
// ---------------------------------------------------------------------------
__global__ void gmm_features(const float* __restrict__ pred_xyz,
                             const float* __restrict__ pred_sigma,
                             const float* __restrict__ target,
                             float* __restrict__ F,
                             float* __restrict__ G) {
    int i = blockIdx.x * blockDim.x + threadIdx.x;
    if (i < BATCH * NPRED) {
        float px = pred_xyz[i * 3 + 0];
        float py = pred_xyz[i * 3 + 1];
        float pz = pred_xyz[i * 3 + 2];
        float sxy = pred_sigma[i * 2 + 0] + EPS_F;
        float sz  = pred_sigma[i * 2 + 1] + EPS_F;
        float a  = 0.5f / sxy;
        float bz = 0.5f / sz;
        float c0 = -__logf(sxy) - 0.5f * __logf(sz) - 1.5f * LOG_2PI_F
                   - a * (px * px + py * py) - bz * pz * pz;
        float* f = F + (size_t)i * 8;
        // permuted: [k0, k1, k4, k5 | k2, k3, k6, k7], all * log2(e)
        f[0] = LOG2E_F * c0;
        f[1] = LOG2E_F * -a;
        f[2] = LOG2E_F * -bz;
        f[3] = LOG2E_F * 2.f * bz * pz;
        f[4] = LOG2E_F * 2.f * a * px;
        f[5] = LOG2E_F * 2.f * a * py;
        f[6] = 0.f;
        f[7] = 0.f;
    }
    if (i < BATCH * MTGT) {
        float tx = target[i * 3 + 0];
        float ty = target[i * 3 + 1];
        float tz = target[i * 3 + 2];
        float* g = G + (size_t)i * 8;
        // permuted: [g0, g1, g4, g5 | g2, g3, g6, g7]
        g[0] = 1.f;
        g[1] = tx * tx + ty * ty;
        g[2] = tz * tz;
        g[3] = tz;
        g[4] = tx;
        g[5] = ty;
        g[6] = 0.f;
        g[7] = 0.f;
    }
}

// ---------------------------------------------------------------------------
// Kernel 2: one workgroup per (batch, 16-target-column tile); 8 waves split
// the N=4096 preds (512 rows each, as 32 tiles of 16 rows).
//
// A-fragment (16x4 f32, ISA 7.12.2): lanes 0-15 row M=lane hold K=0 (v0),
// K=1 (v1); lanes 16-31 same rows hold K=2,K=3. B symmetric. With the
// permuted storage, a lane's fragment pair for both WMMAs is one float4:
//   lanes 0-15 : [k0,k1,k4,k5] at byte offset 0
//   lanes 16-31: [k2,k3,k6,k7] at byte offset 16
// C/D layout: lane holds column N=lane&15; VGPR r = row r (lanes<16) / r+8.
// ---------------------------------------------------------------------------
__global__ __launch_bounds__(256) void gmm_lse(const float* __restrict__ F,
                                               const float* __restrict__ G,
                                               float* __restrict__ nll) {
    const int MT = MTGT / 16;
    int b    = blockIdx.x / MT;
    int mt   = blockIdx.x % MT;
    int lane = threadIdx.x & 31;
    int wave = threadIdx.x >> 5;
    int col  = mt * 16 + (lane & 15);
    int half = (lane < 16) ? 0 : 4;           // float offset into permuted row

    // B fragment (constant across the whole n sweep)
    v4f gv = *(const v4f*)(G + ((size_t)b * MTGT + col) * 8 + half);
    v2f bf0 = { gv.x, gv.y };                 // K = {0,1} or {2,3}
    v2f bf1 = { gv.z, gv.w };                 // K = {4,5} or {6,7}

    float mx = -INFINITY;                     // running max (log2 domain)
    float sm = 0.f;                           // running sum of exp2

    const float* fb = F + (size_t)b * NPRED * 8 + half;
    const int rows_per_wave = NPRED / 8;      // 512
    const int n0 = wave * rows_per_wave;

#pragma unroll 2
    for (int t = 0; t < rows_per_wave; t += 16) {
        v4f fv = *(const v4f*)(fb + (size_t)(n0 + t + (lane & 15)) * 8);
        v2f af0 = { fv.x, fv.y };
        v2f af1 = { fv.z, fv.w };

        v8f acc = {};
        acc = __builtin_amdgcn_wmma_f32_16x16x4_f32(
            false, af0, false, bf0, (short)0, acc, false, false);
        acc = __builtin_amdgcn_wmma_f32_16x16x4_f32(
            false, af1, false, bf1, (short)0, acc, false, false);

        // Online LSE (log2 domain) over this tile's 8 row-values.
        float tm = acc[0];
#pragma unroll
        for (int r = 1; r < 8; ++r) tm = fmaxf(tm, acc[r]);
        float ts = 0.f;
#pragma unroll
        for (int r = 0; r < 8; ++r) ts += fast_exp2(acc[r] - tm);
        float nm = fmaxf(mx, tm);
        sm = sm * fast_exp2(mx - nm) + ts * fast_exp2(tm - nm);
        mx = nm;
    }

    // Merge the two lane halves (rows 0-7 live in lanes 0-15, rows 8-15 in 16-31).
    float om = __shfl_xor(mx, 16, 32);
    float os = __shfl_xor(sm, 16, 32);
    float nm = fmaxf(mx, om);
    sm = sm * fast_exp2(mx - nm) + os * fast_exp2(om - nm);
    mx = nm;

    // Merge across the 8 waves via LDS.
    __shared__ float smax[8][16];
    __shared__ float ssum[8][16];
    if (lane < 16) {
        smax[wave][lane & 15] = mx;
        ssum[wave][lane & 15] = sm;
    }
    __syncthreads();

    if (threadIdx.x < 16) {
        float M0 = smax[0][threadIdx.x];
        float S0 = ssum[0][threadIdx.x];
#pragma unroll
        for (int w = 1; w < 8; ++w) {
            float Mw = smax[w][threadIdx.x];
            float Sw = ssum[w][threadIdx.x];
            float n2 = fmaxf(M0, Mw);
            S0 = S0 * fast_exp2(M0 - n2) + Sw * fast_exp2(Mw - n2);
            M0 = n2;
        }
        // convert log2-domain LSE back to natural log
        float lse = (M0 + fast_log2(S0)) * LN2_F;
        nll[(size_t)b * MTGT + mt * 16 + threadIdx.x] = -lse;
    }
}

// ---------------------------------------------------------------------------
// Kernel 3: mean over the B*M per-target NLL values -> scalar.
// ---------------------------------------------------------------------------
__global__ void gmm_reduce(const float* __restrict__ nll, float* __restrict__ out) {
    __shared__ float buf[256];
    float s = 0.f;
    for (int i = threadIdx.x; i < BATCH * MTGT; i += 256) s += nll[i];
    buf[threadIdx.x] = s;
    __syncthreads();
    for (int off = 128; off > 0; off >>= 1) {
        if ((int)threadIdx.x < off) buf[threadIdx.x] += buf[threadIdx.x + off];
        __syncthreads();
    }
    if (threadIdx.x == 0) out[0] = buf[0] / (float)(BATCH * MTGT);
}

// ---------------------------------------------------------------------------
extern "C" void kernel_launch(void* const* d_in, const int* in_sizes, int n_in,
                              void* d_out, int out_size, void* d_ws, size_t ws_size,
                              hipStream_t stream) {
    const float* pred_xyz   = (const float*)d_in[0];  // (B, N, 3)
    const float* pred_sigma = (const float*)d_in[1];  // (B, N, 2)
    const float* target     = (const float*)d_in[2];  // (B, M, 3)

    float* ws  = (float*)d_ws;
    float* F   = ws;                               // B*N*8 floats (512 KB)
    float* G   = F + (size_t)BATCH * NPRED * 8;    // B*M*8 floats (512 KB)
    float* nll = G + (size_t)BATCH * MTGT * 8;     // B*M floats   (64 KB)

    int nmax = BATCH * (NPRED > MTGT ? NPRED : MTGT);
    gmm_features<<<(nmax + 255) / 256, 256, 0, stream>>>(pred_xyz, pred_sigma,
                                                         target, F, G);
    gmm_lse<<<BATCH * (MTGT / 16), 256, 0, stream>>>(F, G, nll);
    gmm_reduce<<<1, 256, 0, stream>>>(nll, (float*)d_out);
}